// GraphSingleAttentionStream_57947698758300
// MI455X (gfx1250) — compile-verified
//
#include <hip/hip_runtime.h>
#include <hip/hip_bf16.h>
#include <math.h>

#define N_NODES_C 100000
#define N_EDGES_C 1000000
#define NFEAT_C   256
#define NHID_C    64
#define ALPHA_C   0.2f

typedef __attribute__((ext_vector_type(2))) float v2f;
typedef __attribute__((ext_vector_type(8))) float v8f;

static __device__ __forceinline__ v8f wmma_f32_16x16x4(v2f a, v2f b, v8f c) {
  // 8 args: (neg_a, A, neg_b, B, c_mod, C, reuse_a, reuse_b)
  return __builtin_amdgcn_wmma_f32_16x16x4_f32(false, a, false, b, (short)0, c,
                                               false, false);
}

// ---------------------------------------------------------------------------
// Phase 0: fold the attention matvecs:
//   c_src  = W_f @ a_src   (256)      off_src  = b_f . a_src
//   c_dest = W_f @ a_dest  (256)      off_dest = b_f . a_dest
// so that f1 = feat @ c_src + off_src (eliminates the whole h = feat@W_f GEMM
// and a second 102 MB pass over feat).
// ---------------------------------------------------------------------------
__global__ __launch_bounds__(256) void precompute_c(
    const float* __restrict__ Wf, const float* __restrict__ bf,
    const float* __restrict__ a_src, const float* __restrict__ a_dest,
    float* __restrict__ c_src, float* __restrict__ c_dest,
    float* __restrict__ offs) {
  int k = threadIdx.x;  // 256 threads, one output each
  const float* wr = Wf + k * NHID_C;
  float s1 = 0.f, s2 = 0.f;
  for (int j = 0; j < NHID_C; ++j) {
    float w = wr[j];
    s1 += w * a_src[j];
    s2 += w * a_dest[j];
  }
  c_src[k] = s1;
  c_dest[k] = s2;
  if (k == 0) {
    float o1 = 0.f, o2 = 0.f;
    for (int j = 0; j < NHID_C; ++j) {
      o1 += bf[j] * a_src[j];
      o2 += bf[j] * a_dest[j];
    }
    offs[0] = o1;
    offs[1] = o2;
  }
}

// ---------------------------------------------------------------------------
// WMMA GEMM: Y[N,64] = X[N,K] @ W[K,64] + bias, fp32 WMMA 16x16x4.
// 8 waves/block, each wave owns 16 rows x 64 cols (4 v8f accumulators).
// W is staged in LDS in K-blocks of 64 using a k-pair-interleaved layout:
//   ldsW[(k/2)*128 + 2n + (k&1)] = W[k][n]
// so each B fragment (W[kb][n], W[kb+1][n]) is ONE aligned ds_load_b64
// straight into an even VGPR pair (no register reassembly movs).
// Optionally fuses the f1/f2 matvecs while A fragments are in registers.
// ---------------------------------------------------------------------------
template <int K, bool FUSE_F>
__global__ __launch_bounds__(256) void gemm_wmma(
    const float* __restrict__ X, const float* __restrict__ W,
    const float* __restrict__ bias, const float* __restrict__ cvec,
    const float* __restrict__ offs, float* __restrict__ Y,
    float* __restrict__ f1, float* __restrict__ f2, int nrows) {
  constexpr int KBLK = 64;
  __shared__ float ldsW[KBLK * 64];  // pair-interleaved, 16 KB
  __shared__ float ldsC[2 * K];

  if (FUSE_F) {
    for (int i = threadIdx.x; i < 2 * K; i += 256) ldsC[i] = cvec[i];
  }

  const int wave = threadIdx.x >> 5;
  const int lane = threadIdx.x & 31;
  const int half = lane >> 4;    // which 16-lane half
  const int ncol = lane & 15;    // column within 16-wide tile / row within A
  const int rowbase = blockIdx.x * 128 + wave * 16;

  // A-fragment source row for this lane (clamped; OOB rows never stored)
  int m = rowbase + ncol;
  if (m >= nrows) m = nrows - 1;
  const float* xr = X + (long)m * K;

  v8f acc[4] = {};
  float p1 = 0.f, p2 = 0.f;

  for (int kk = 0; kk < K; kk += KBLK) {
    __syncthreads();
    // coalesced global read, pair-interleaved LDS write
    for (int i = threadIdx.x; i < KBLK * 64; i += 256) {
      const int k = i >> 6, n = i & 63;
      ldsW[(k >> 1) * 128 + n * 2 + (k & 1)] = W[(kk + k) * 64 + n];
    }
    __syncthreads();

#pragma unroll
    for (int k0 = 0; k0 < KBLK; k0 += 4) {
      const int kb = k0 + half * 2;  // lanes 0-15: k0,k0+1; lanes 16-31: k0+2,k0+3
      const v2f a = *(const v2f*)(xr + kk + kb);
      if (FUSE_F) {
        const v2f cs = *(const v2f*)&ldsC[kk + kb];
        const v2f cd = *(const v2f*)&ldsC[K + kk + kb];
        p1 += a.x * cs.x + a.y * cs.y;
        p2 += a.x * cd.x + a.y * cd.y;
      }
      const float* wp = &ldsW[(kb >> 1) * 128];
      acc[0] = wmma_f32_16x16x4(a, *(const v2f*)&wp[(ncol)      * 2], acc[0]);
      acc[1] = wmma_f32_16x16x4(a, *(const v2f*)&wp[(16 + ncol) * 2], acc[1]);
      acc[2] = wmma_f32_16x16x4(a, *(const v2f*)&wp[(32 + ncol) * 2], acc[2]);
      acc[3] = wmma_f32_16x16x4(a, *(const v2f*)&wp[(48 + ncol) * 2], acc[3]);
    }
  }

  // C/D layout: lanes 0-15 -> M = r, lanes 16-31 -> M = 8 + r (VGPR r), N = ncol
#pragma unroll
  for (int t = 0; t < 4; ++t) {
    const float bv = bias[t * 16 + ncol];
#pragma unroll
    for (int r = 0; r < 8; ++r) {
      const int mr = rowbase + half * 8 + r;
      if (mr < nrows) Y[(long)mr * 64 + t * 16 + ncol] = acc[t][r] + bv;
    }
  }

  if (FUSE_F) {
    // lanes L and L+16 hold the k%4 in {0,1} and {2,3} partials of row L
    p1 += __shfl_xor(p1, 16, 32);
    p2 += __shfl_xor(p2, 16, 32);
    const int mr = rowbase + ncol;
    if (half == 0 && mr < nrows) {
      f1[mr] = p1 + offs[0];
      f2[mr] = p2 + offs[1];
    }
  }
}

// ---------------------------------------------------------------------------
// row_ptr[t] = lower_bound(row, t) over the sorted COO row array.
// ---------------------------------------------------------------------------
__global__ __launch_bounds__(256) void build_rowptr(const int* __restrict__ row,
                                                    int* __restrict__ rp) {
  int t = blockIdx.x * blockDim.x + threadIdx.x;
  if (t > N_NODES_C) return;
  int lo = 0, hi = N_EDGES_C;
  while (lo < hi) {
    int mid = (lo + hi) >> 1;
    if (row[mid] < t) lo = mid + 1; else hi = mid;
  }
  rp[t] = lo;
}

// ---------------------------------------------------------------------------
// Per-row edge softmax (one wave32 per row; no atomics):
//   e = leakyrelu(f1[r] + f2[col]); att = exp(e - max)/sum
// ---------------------------------------------------------------------------
__global__ __launch_bounds__(256) void edge_softmax(
    const int* __restrict__ rp, const int* __restrict__ col,
    const float* __restrict__ f1, const float* __restrict__ f2,
    float* __restrict__ att) {
  const int r = blockIdx.x * 8 + (threadIdx.x >> 5);
  const int lane = threadIdx.x & 31;
  if (r >= N_NODES_C) return;
  const int beg = rp[r], end = rp[r + 1];
  const float f1r = f1[r];

  float mx = -INFINITY;
  for (int j = beg + lane; j < end; j += 32) {
    float v = f1r + f2[col[j]];
    float e = v > 0.f ? v : ALPHA_C * v;
    mx = fmaxf(mx, e);
  }
  for (int o = 16; o > 0; o >>= 1) mx = fmaxf(mx, __shfl_xor(mx, o, 32));

  float s = 0.f;
  for (int j = beg + lane; j < end; j += 32) {
    float v = f1r + f2[col[j]];
    float e = v > 0.f ? v : ALPHA_C * v;
    float ex = __expf(e - mx);
    att[j] = ex;
    s += ex;
  }
  for (int o = 16; o > 0; o >>= 1) s += __shfl_xor(s, o, 32);

  const float inv = (end > beg) ? 1.f / s : 0.f;
  for (int j = beg + lane; j < end; j += 32) att[j] *= inv;
}

// ---------------------------------------------------------------------------
// SpMM + ELU, one wave32 per row: Y[r,:] = ELU( sum_j att[j] * X[col[j],:] )
// Each lane owns 2 of the 64 features (float2 = B64 loads).
// ---------------------------------------------------------------------------
__global__ __launch_bounds__(256) void spmm_elu(
    const int* __restrict__ rp, const int* __restrict__ col,
    const float* __restrict__ att, const float* __restrict__ X,
    float* __restrict__ Y) {
  const int r = blockIdx.x * 8 + (threadIdx.x >> 5);
  const int lane = threadIdx.x & 31;
  if (r >= N_NODES_C) return;
  const int beg = rp[r], end = rp[r + 1];

  float a0 = 0.f, a1 = 0.f;
  for (int j = beg; j < end; ++j) {
    const float a = att[j];
    const float2 xc = *(const float2*)(X + (long)col[j] * 64 + lane * 2);
    a0 += a * xc.x;
    a1 += a * xc.y;
  }
  a0 = a0 > 0.f ? a0 : __expf(a0) - 1.f;
  a1 = a1 > 0.f ? a1 : __expf(a1) - 1.f;
  float2* y = (float2*)(Y + (long)r * 64 + lane * 2);
  *y = make_float2(a0, a1);
}

// ---------------------------------------------------------------------------
extern "C" void kernel_launch(void* const* d_in, const int* in_sizes, int n_in,
                              void* d_out, int out_size, void* d_ws,
                              size_t ws_size, hipStream_t stream) {
  const float* feat  = (const float*)d_in[0];
  const int*   row   = (const int*)d_in[1];
  const int*   col   = (const int*)d_in[2];
  const float* W_f   = (const float*)d_in[3];
  const float* b_f   = (const float*)d_in[4];
  const float* a_src = (const float*)d_in[5];
  const float* a_dst = (const float*)d_in[6];
  const float* W0    = (const float*)d_in[7];
  const float* b0    = (const float*)d_in[8];
  const float* W1    = (const float*)d_in[9];
  const float* b1    = (const float*)d_in[10];
  float* out = (float*)d_out;

  char* ws = (char*)d_ws;
  size_t off = 0;
  auto alloc = [&](size_t bytes) -> void* {
    void* p = ws + off;
    off = (off + bytes + 255) & ~(size_t)255;
    return p;
  };

  float* cvec = (float*)alloc(2 * NFEAT_C * sizeof(float));  // c_src | c_dest
  float* offs = (float*)alloc(2 * sizeof(float));
  float* f1   = (float*)alloc((size_t)N_NODES_C * sizeof(float));
  float* f2   = (float*)alloc((size_t)N_NODES_C * sizeof(float));
  float* xw   = (float*)alloc((size_t)N_NODES_C * 64 * sizeof(float));  // GEMM0 out, reused for GEMM1 out
  float* x1   = (float*)alloc((size_t)N_NODES_C * 64 * sizeof(float));  // layer0 activations
  float* att  = (float*)alloc((size_t)N_EDGES_C * sizeof(float));
  int*   rp   = (int*)alloc((size_t)(N_NODES_C + 1) * sizeof(int));

  const int gemm_blocks = (N_NODES_C + 127) / 128;     // 782
  const int row_blocks  = (N_NODES_C + 7) / 8;         // 12500
  const int rp_blocks   = (N_NODES_C + 1 + 255) / 256; // 391

  // 0: fold attention matvecs
  precompute_c<<<1, 256, 0, stream>>>(W_f, b_f, a_src, a_dst, cvec,
                                      cvec + NFEAT_C, offs);
  // 1: xw = feat @ W0 + b0  (WMMA), fused f1/f2
  gemm_wmma<NFEAT_C, true><<<gemm_blocks, 256, 0, stream>>>(
      feat, W0, b0, cvec, offs, xw, f1, f2, N_NODES_C);
  // 2: CSR row pointers from sorted COO rows
  build_rowptr<<<rp_blocks, 256, 0, stream>>>(row, rp);
  // 3: per-row edge softmax -> att
  edge_softmax<<<row_blocks, 256, 0, stream>>>(rp, col, f1, f2, att);
  // 4: x1 = ELU(SpMM(att, xw))
  spmm_elu<<<row_blocks, 256, 0, stream>>>(rp, col, att, xw, x1);
  // 5: xw = x1 @ W1 + b1 (WMMA)
  gemm_wmma<NHID_C, false><<<gemm_blocks, 256, 0, stream>>>(
      x1, W1, b1, nullptr, nullptr, xw, nullptr, nullptr, N_NODES_C);
  // 6: out = ELU(SpMM(att, xw))
  spmm_elu<<<row_blocks, 256, 0, stream>>>(rp, col, att, xw, out);
}